// Quantizout_24816321036973
// MI455X (gfx1250) — compile-verified
//
#include <hip/hip_runtime.h>
#include <hip/hip_bf16.h>

// out[i] = (noise[i] < PROB) ? rint(x[i]) : x[i]   (PROB = 0.5, BINARY=False)
//
// Pure streaming elementwise op: 616 MB of traffic, ~0.1 GFLOP.
// Bandwidth-bound on MI455X (23.3 TB/s -> ~26 us floor). Strategy:
//   - float4 (b128) loads/stores, 512B per wave32 transaction
//   - non-temporal hints (TH=NT): 616 MB working set >> 192 MB L2, zero reuse
//   - rintf -> v_rndne_f32 (round-half-to-even, matches jnp.round)
//   - wave32-friendly 256-thread blocks, exact-cover flat launch

typedef float v4f __attribute__((ext_vector_type(4)));

__global__ __launch_bounds__(256) void
Quantizout_24816321036973_kernel(const float* __restrict__ x,
                                 const float* __restrict__ noise,
                                 float* __restrict__ out,
                                 long long nvec,   // number of float4 quads
                                 long long n)      // total elements
{
    const long long i = (long long)blockIdx.x * blockDim.x + threadIdx.x;

    if (i < nvec) {
        const v4f* __restrict__ x4 = (const v4f*)x;
        const v4f* __restrict__ n4 = (const v4f*)noise;
        v4f*       __restrict__ o4 = (v4f*)out;

        // NT loads: stream through L2 without displacing resident lines
        v4f xv = __builtin_nontemporal_load(&x4[i]);
        v4f nz = __builtin_nontemporal_load(&n4[i]);

        v4f r;
        r.x = (nz.x < 0.5f) ? rintf(xv.x) : xv.x;
        r.y = (nz.y < 0.5f) ? rintf(xv.y) : xv.y;
        r.z = (nz.z < 0.5f) ? rintf(xv.z) : xv.z;
        r.w = (nz.w < 0.5f) ? rintf(xv.w) : xv.w;

        __builtin_nontemporal_store(r, &o4[i]);
    }

    // Scalar tail (n % 4 == 0 for the 64*256*56*56 case, kept for generality).
    const long long tail_base = nvec * 4;
    const long long rem = n - tail_base;   // 0..3
    if (i < rem) {
        const long long j = tail_base + i;
        float xv = __builtin_nontemporal_load(&x[j]);
        float nz = __builtin_nontemporal_load(&noise[j]);
        float r  = (nz < 0.5f) ? rintf(xv) : xv;
        __builtin_nontemporal_store(r, &out[j]);
    }
}

extern "C" void kernel_launch(void* const* d_in, const int* in_sizes, int n_in,
                              void* d_out, int out_size, void* d_ws, size_t ws_size,
                              hipStream_t stream) {
    const float* x     = (const float*)d_in[0];
    const float* noise = (const float*)d_in[1];
    float*       out   = (float*)d_out;

    const long long n    = (long long)in_sizes[0];  // 51,380,224
    const long long nvec = n >> 2;                  // float4 quads

    const int block = 256;                          // 8 wave32 waves
    const long long work = (nvec > 0) ? nvec : 1;   // ensure tail threads exist
    const long long grid = (work + block - 1) / block;

    Quantizout_24816321036973_kernel<<<(dim3)(unsigned)grid, block, 0, stream>>>(
        x, noise, out, nvec, n);
}